// RetrievalBranch_17746804867780
// MI455X (gfx1250) — compile-verified
//
#include <hip/hip_runtime.h>
#include <math.h>
#include <stdint.h>

// ---------------------------------------------------------------------------
// RetrievalBranch for MI455X (gfx1250, wave32, WMMA + async LDS copies)
//   score = 0.5 * (norm(z_ego) @ norm(exo_bank)^T) + 0.5 * sim_vt
//   topk_idx = top-20 per row; z_exo = exo_bank[topk_idx]
// FP32 WMMA (16x16x4): top-k gaps (~1e-5) are below bf16/f16 dot error.
// Global->LDS staging uses GLOBAL_LOAD_ASYNC_TO_LDS_B128 (ASYNCcnt) with
// double-buffered LDS so the 410MB exo stream overlaps WMMA compute.
// ---------------------------------------------------------------------------

typedef float v2f __attribute__((ext_vector_type(2)));
typedef float v8f __attribute__((ext_vector_type(8)));

#define TPB 256
#define NTILE 128          // exo rows per block
#define KC 64              // K chunk staged in LDS
#define LSTR 72            // LDS row stride in floats (288B: 16B aligned, bank-spread)

// ---------------- kernel 1: normalize z_ego rows -> zn ---------------------
__global__ __launch_bounds__(TPB)
void row_normalize_kernel(const float* __restrict__ x, float* __restrict__ y, int D) {
    __shared__ float sdata[TPB];
    const int row = blockIdx.x, tid = threadIdx.x;
    const float4* xr = (const float4*)(x + (size_t)row * D);
    float s = 0.f;
    for (int i = tid; i < (D >> 2); i += TPB) {
        float4 v = xr[i];
        s += v.x * v.x + v.y * v.y + v.z * v.z + v.w * v.w;
    }
    sdata[tid] = s;
    __syncthreads();
    for (int st = TPB / 2; st > 0; st >>= 1) {
        if (tid < st) sdata[tid] += sdata[tid + st];
        __syncthreads();
    }
    const float inv = 1.0f / fmaxf(sqrtf(sdata[0]), 1e-12f);
    float4* yr = (float4*)(y + (size_t)row * D);
    for (int i = tid; i < (D >> 2); i += TPB) {
        float4 v = xr[i];
        v.x *= inv; v.y *= inv; v.z *= inv; v.w *= inv;
        yr[i] = v;
    }
}

// ---------------- kernel 2: inverse norm of each exo row -------------------
__global__ __launch_bounds__(TPB)
void row_invnorm_kernel(const float* __restrict__ x, float* __restrict__ invn, int D) {
    __shared__ float sdata[TPB];
    const int row = blockIdx.x, tid = threadIdx.x;
    const float4* xr = (const float4*)(x + (size_t)row * D);
    float s = 0.f;
    for (int i = tid; i < (D >> 2); i += TPB) {
        float4 v = xr[i];
        s += v.x * v.x + v.y * v.y + v.z * v.z + v.w * v.w;
    }
    sdata[tid] = s;
    __syncthreads();
    for (int st = TPB / 2; st > 0; st >>= 1) {
        if (tid < st) sdata[tid] += sdata[tid + st];
        __syncthreads();
    }
    if (tid == 0) invn[row] = 1.0f / fmaxf(sqrtf(sdata[0]), 1e-12f);
}

// -------- async global->LDS staging of one K-chunk (ASYNCcnt tracked) ------
// Each of 256 threads issues 4 A-tile and 8 B-tile 16B async copies.
// VDST operand = per-lane LDS byte address (low 32 bits of generic pointer),
// VADDR = 64-bit global address, GV mode ("off").
__device__ __forceinline__ void async_stage_chunk(
    const float* __restrict__ zn, const float* __restrict__ exo,
    float* ldsA, float* ldsB, int tid, int n0, int kbase, int N, int D, int B) {
#pragma unroll
    for (int it = 0; it < 4; ++it) {           // A: 64 rows x 16 float4
        const int i = it * TPB + tid;
        const int row = i >> 4, c4 = i & 15;
        const int rr = (row < B) ? row : (B - 1);
        const float* g = zn + (size_t)rr * D + kbase + c4 * 4;
        const unsigned l = (unsigned)(uintptr_t)(ldsA + row * LSTR + c4 * 4);
        asm volatile("global_load_async_to_lds_b128 %0, %1, off"
                     :: "v"(l), "v"(g) : "memory");
    }
#pragma unroll
    for (int it = 0; it < 8; ++it) {           // B: 128 rows x 16 float4
        const int i = it * TPB + tid;
        const int r = i >> 4, c4 = i & 15;
        int n = n0 + r; if (n >= N) n = N - 1; // clamp; garbage cols never stored
        const float* g = exo + (size_t)n * D + kbase + c4 * 4;
        const unsigned l = (unsigned)(uintptr_t)(ldsB + r * LSTR + c4 * 4);
        asm volatile("global_load_async_to_lds_b128 %0, %1, off"
                     :: "v"(l), "v"(g) : "memory");
    }
}

// ---------------- kernel 3: WMMA GEMM + score epilogue ---------------------
// Block: 256 threads (8 waves). Each block: NTILE=128 exo rows x M=64 ego rows.
// Wave w owns N-subtile [w*16, w*16+16) and all 4 M-subtiles of 16.
// A (16x4 f32, ISA 7.12.2): lane m=l&15, half h=l>>4: VGPR0=A[m][k+2h], VGPR1=A[m][k+2h+1]
// B (4x16 f32): lane n=l&15, half h: VGPR0=B[k+h][n], VGPR1=B[k+2+h][n]; B[k][n]=exo[n][k]
// C/D (16x16 f32): lane n=l&15; VGPR v -> m = v + 8h
__global__ __launch_bounds__(TPB)
void score_wmma_kernel(const float* __restrict__ zn, const float* __restrict__ exo,
                       const float* __restrict__ invn, const float* __restrict__ simvt,
                       float* __restrict__ score, int N, int D, int B) {
    __shared__ float ldsA[2][64 * LSTR];      // 2 x 18.0 KB
    __shared__ float ldsB[2][NTILE * LSTR];   // 2 x 36.9 KB  (total ~110 KB)

    const int tid  = threadIdx.x;
    const int wave = tid >> 5;
    const int lane = tid & 31;
    const int l15  = lane & 15;
    const int h    = lane >> 4;
    const int n0   = blockIdx.x * NTILE;

    v8f acc0 = {}, acc1 = {}, acc2 = {}, acc3 = {};

    const int nchunks = D / KC;
    async_stage_chunk(zn, exo, ldsA[0], ldsB[0], tid, n0, 0, N, D, B);

    for (int c = 0; c < nchunks; ++c) {
        // my async copies for chunk c are done; barrier => everyone's are,
        // and everyone finished reading buffer (c+1)&1 in iteration c-1.
        asm volatile("s_wait_asynccnt 0x0" ::: "memory");
        __syncthreads();
        if (c + 1 < nchunks)
            async_stage_chunk(zn, exo, ldsA[(c + 1) & 1], ldsB[(c + 1) & 1],
                              tid, n0, (c + 1) * KC, N, D, B);

        const float* Bp = ldsB[c & 1] + (wave * 16 + l15) * LSTR;
        const float* Ap = ldsA[c & 1] + l15 * LSTR;
#pragma unroll
        for (int kk = 0; kk < KC; kk += 4) {
            v2f bf;
            bf.x = Bp[kk + h];
            bf.y = Bp[kk + 2 + h];
            v2f a;
            a.x = Ap[kk + 2 * h];
            a.y = Ap[kk + 2 * h + 1];
            acc0 = __builtin_amdgcn_wmma_f32_16x16x4_f32(false, a, false, bf, (short)0, acc0, false, false);
            a.x = Ap[16 * LSTR + kk + 2 * h];
            a.y = Ap[16 * LSTR + kk + 2 * h + 1];
            acc1 = __builtin_amdgcn_wmma_f32_16x16x4_f32(false, a, false, bf, (short)0, acc1, false, false);
            a.x = Ap[32 * LSTR + kk + 2 * h];
            a.y = Ap[32 * LSTR + kk + 2 * h + 1];
            acc2 = __builtin_amdgcn_wmma_f32_16x16x4_f32(false, a, false, bf, (short)0, acc2, false, false);
            a.x = Ap[48 * LSTR + kk + 2 * h];
            a.y = Ap[48 * LSTR + kk + 2 * h + 1];
            acc3 = __builtin_amdgcn_wmma_f32_16x16x4_f32(false, a, false, bf, (short)0, acc3, false, false);
        }
    }

    // Epilogue: score[m][n] = 0.5 * dot * invn[n] + 0.5 * simvt[n]
    const int n = n0 + wave * 16 + l15;
    if (n < N) {
        const float scale = 0.5f * invn[n];
        const float add   = 0.5f * simvt[n];
#pragma unroll
        for (int v = 0; v < 8; ++v) {
            const int m = v + 8 * h;
            if (m      < B) score[(size_t)(m)      * N + n] = acc0[v] * scale + add;
            if (m + 16 < B) score[(size_t)(m + 16) * N + n] = acc1[v] * scale + add;
            if (m + 32 < B) score[(size_t)(m + 32) * N + n] = acc2[v] * scale + add;
            if (m + 48 < B) score[(size_t)(m + 48) * N + n] = acc3[v] * scale + add;
        }
    }
}

// ---------------- kernel 4: top-k per row (iterative argmax) ---------------
__global__ __launch_bounds__(TPB)
void topk_kernel(const float* __restrict__ score, int N, int k_eff,
                 int* __restrict__ out_idx) {
    __shared__ float rv[TPB];
    __shared__ int   ri[TPB];
    __shared__ int   sel[32];
    const int b = blockIdx.x, tid = threadIdx.x;
    const float* row = score + (size_t)b * N;

    for (int j = 0; j < k_eff; ++j) {
        float bv = -INFINITY;
        int   bi = 0x7fffffff;
        for (int i = tid; i < N; i += TPB) {
            bool taken = false;
            for (int t = 0; t < j; ++t) taken |= (sel[t] == i);
            const float v = row[i];
            if (!taken && (v > bv || (v == bv && i < bi))) { bv = v; bi = i; }
        }
        rv[tid] = bv; ri[tid] = bi;
        __syncthreads();
        for (int st = TPB / 2; st > 0; st >>= 1) {
            if (tid < st) {
                const float v2 = rv[tid + st]; const int i2 = ri[tid + st];
                if (v2 > rv[tid] || (v2 == rv[tid] && i2 < ri[tid])) {
                    rv[tid] = v2; ri[tid] = i2;
                }
            }
            __syncthreads();
        }
        if (tid == 0) { sel[j] = ri[0]; out_idx[b * k_eff + j] = ri[0]; }
        __syncthreads();
    }
}

// ---------------- kernel 5: gather z_exo = exo_bank[idx] -------------------
__global__ __launch_bounds__(TPB)
void gather_kernel(const float* __restrict__ exo, const int* __restrict__ idx,
                   float* __restrict__ out, int D) {
    const int bj = blockIdx.x;
    const int id = idx[bj];
    const float4* src = (const float4*)(exo + (size_t)id * D);
    float4*       dst = (float4*)(out + (size_t)bj * D);
    for (int i = threadIdx.x; i < (D >> 2); i += TPB) dst[i] = src[i];
}

// ---------------------------------------------------------------------------
extern "C" void kernel_launch(void* const* d_in, const int* in_sizes, int n_in,
                              void* d_out, int out_size, void* d_ws, size_t ws_size,
                              hipStream_t stream) {
    const float* z_ego = (const float*)d_in[0];   // [B, D]
    const float* exo   = (const float*)d_in[1];   // [N, D]
    // d_in[2] = k (device scalar); k_eff is recovered from out_size below
    const float* simvt = (const float*)d_in[3];   // [N]

    const int N = in_sizes[3];
    const int D = in_sizes[1] / N;
    const int B = in_sizes[0] / D;
    const int k_eff = out_size / (B * (D + 1));   // out = B*k*D floats + B*k ints

    // Workspace layout (floats)
    float* zn    = (float*)d_ws;                  // B * D
    float* invn  = zn + (size_t)B * D;            // N
    float* score = invn + N;                      // B * N  (~25.6 MB)

    float* z_out   = (float*)d_out;                              // [B, k, D]
    int*   idx_out = (int*)(z_out + (size_t)B * k_eff * D);      // [B, k] int32 bits

    row_normalize_kernel<<<B, TPB, 0, stream>>>(z_ego, zn, D);
    row_invnorm_kernel<<<N, TPB, 0, stream>>>(exo, invn, D);

    const int nblocks = (N + NTILE - 1) / NTILE;
    score_wmma_kernel<<<nblocks, TPB, 0, stream>>>(zn, exo, invn, simvt, score, N, D, B);

    topk_kernel<<<B, TPB, 0, stream>>>(score, N, k_eff, idx_out);
    gather_kernel<<<B * k_eff, TPB, 0, stream>>>(exo, idx_out, z_out, D);
}